// Bootstrap_Proposal_52785148067878
// MI455X (gfx1250) — compile-verified
//
#include <hip/hip_runtime.h>
#include <hip/hip_bf16.h>
#include <stdint.h>

// Bootstrap_Proposal mean update (time != 0 branch), L1=L2=M1=M2=1, DT=0.01.
// Per row [tq1, tq2, th1, th2, v1, v2]:
//   c    = cos(th2)
//   d00  = 5/3 + c          (== (5+3c)/3)
//   d01  = c/2 + 1/3        (== (3c+2)/6)
//   d11  = 1/3
//   det  = (16 - 9c^2)/36   -> inv_det = 36 / (16 - 9c^2)
//   a1   = (d11*tq1 - d01*tq2) * inv_det
//   a2   = (d00*tq2 - d01*tq1) * inv_det
//   out  = [tq1, tq2, th1+DT*v1, th2+DT*v2, v1+DT*a1, v2+DT*a2]
//
// Bandwidth-bound: 96 MB total traffic -> ~4.1 us floor at 23.3 TB/s.
// Strategy: stage 12 KB tiles (512 rows) per 256-thread block through LDS
// with gfx1250 async b128 global<->LDS DMA (ASYNCcnt), compute in place in
// LDS (conflict-free: stride-6-dword rows across wave32 lanes), stream back.

#define THREADS 256
#define ROWS_PER_BLOCK 512                    // 512 rows * 24 B = 12288 B tile
#define TILE_FLOATS (ROWS_PER_BLOCK * 6)      // 3072 floats
#define CHUNKS_PER_THREAD 3                   // 12288 B / 16 B / 256 thr

__device__ __forceinline__ void row_update(float* row) {
    const float tq1 = row[0], tq2 = row[1];
    const float th1 = row[2], th2 = row[3];
    const float v1  = row[4], v2  = row[5];

    const float c     = __cosf(th2);                            // v_cos_f32
    const float d01   = __builtin_fmaf(0.5f, c, 1.0f / 3.0f);   // (3c+2)/6
    const float d00   = c + (5.0f / 3.0f);                      // (5+3c)/3
    const float det36 = __builtin_fmaf(-9.0f * c, c, 16.0f);    // 36*det
    const float inv   = 36.0f * __builtin_amdgcn_rcpf(det36);   // v_rcp_f32

    const float a1 = __builtin_fmaf(1.0f / 3.0f, tq1, -d01 * tq2) * inv;
    const float a2 = __builtin_fmaf(d00, tq2, -d01 * tq1) * inv;

    float2 o23 = make_float2(__builtin_fmaf(0.01f, v1, th1),
                             __builtin_fmaf(0.01f, v2, th2));
    float2 o45 = make_float2(__builtin_fmaf(0.01f, a1, v1),
                             __builtin_fmaf(0.01f, a2, v2));
    // rows are 24 B apart -> offsets +8 and +16 are 8-byte aligned
    *(float2*)(row + 2) = o23;   // ds_store_b64
    *(float2*)(row + 4) = o45;   // ds_store_b64
}

__global__ __launch_bounds__(THREADS) void
bootstrap_mean_tile_kernel(const float* __restrict__ in, float* __restrict__ out)
{
    __shared__ float tile[TILE_FLOATS];

    const unsigned tid = threadIdx.x;
    const float* gsrc = in  + (size_t)blockIdx.x * TILE_FLOATS;  // uniform -> SGPR pair
    float*       gdst = out + (size_t)blockIdx.x * TILE_FLOATS;

    // Low 32 bits of a flat shared pointer == byte offset within workgroup LDS.
    const unsigned lds_base = (unsigned)(uintptr_t)(&tile[0]);

    // ---- Phase 1: async DMA global -> LDS, 3 coalesced b128 per thread ----
#pragma unroll
    for (int i = 0; i < CHUNKS_PER_THREAD; ++i) {
        const unsigned byteoff  = (tid + i * THREADS) * 16u;
        const unsigned lds_addr = lds_base + byteoff;
        asm volatile("global_load_async_to_lds_b128 %0, %1, %2"
                     :: "v"(lds_addr), "v"(byteoff), "s"(gsrc)
                     : "memory");
    }
    asm volatile("s_wait_asynccnt 0" ::: "memory");  // my ASYNCcnt drained
    __syncthreads();                                  // everyone's tile visible

    // ---- Phase 2: compute 2 rows per thread, in place in LDS ----
    // lane L touches dwords 6L.. : 6*dL % 64 == 0 only for dL % 32 == 0
    // -> zero bank conflicts within a wave32.
#pragma unroll
    for (int i = 0; i < 2; ++i) {
        const unsigned r = tid + i * THREADS;
        row_update(&tile[r * 6]);
    }
    __syncthreads();

    // ---- Phase 3: async DMA LDS -> global, 3 coalesced b128 per thread ----
#pragma unroll
    for (int i = 0; i < CHUNKS_PER_THREAD; ++i) {
        const unsigned byteoff  = (tid + i * THREADS) * 16u;
        const unsigned lds_addr = lds_base + byteoff;
        asm volatile("global_store_async_from_lds_b128 %0, %1, %2"
                     :: "v"(byteoff), "v"(lds_addr), "s"(gdst)
                     : "memory");
    }
    asm volatile("s_wait_asynccnt 0" ::: "memory");  // (s_endpgm also waits idle)
}

// Defensive tail for row counts not divisible by ROWS_PER_BLOCK.
// Rows start at 24-byte offsets -> 8-byte aligned -> float2 loads are legal.
__global__ __launch_bounds__(THREADS) void
bootstrap_mean_tail_kernel(const float* __restrict__ in, float* __restrict__ out,
                           long long start_row, long long n_rows)
{
    const long long r = start_row + (long long)blockIdx.x * THREADS + threadIdx.x;
    if (r >= n_rows) return;

    const float* src = in  + r * 6;
    float*       dst = out + r * 6;

    float2 p01 = *(const float2*)(src + 0);
    float2 p23 = *(const float2*)(src + 2);
    float2 p45 = *(const float2*)(src + 4);

    const float c     = __cosf(p23.y);
    const float d01   = __builtin_fmaf(0.5f, c, 1.0f / 3.0f);
    const float d00   = c + (5.0f / 3.0f);
    const float det36 = __builtin_fmaf(-9.0f * c, c, 16.0f);
    const float inv   = 36.0f * __builtin_amdgcn_rcpf(det36);

    const float a1 = __builtin_fmaf(1.0f / 3.0f, p01.x, -d01 * p01.y) * inv;
    const float a2 = __builtin_fmaf(d00, p01.y, -d01 * p01.x) * inv;

    float2 o23 = make_float2(__builtin_fmaf(0.01f, p45.x, p23.x),
                             __builtin_fmaf(0.01f, p45.y, p23.y));
    float2 o45 = make_float2(__builtin_fmaf(0.01f, a1, p45.x),
                             __builtin_fmaf(0.01f, a2, p45.y));

    *(float2*)(dst + 0) = p01;
    *(float2*)(dst + 2) = o23;
    *(float2*)(dst + 4) = o45;
}

extern "C" void kernel_launch(void* const* d_in, const int* in_sizes, int n_in,
                              void* d_out, int out_size, void* d_ws, size_t ws_size,
                              hipStream_t stream)
{
    (void)in_sizes; (void)n_in; (void)d_ws; (void)ws_size;
    const float* in  = (const float*)d_in[0];   // prev_latents [B, P, 6] f32
    float*       out = (float*)d_out;           // mean         [B, P, 6] f32

    const long long rows        = (long long)out_size / 6;            // 2,097,152
    const long long full_blocks = rows / ROWS_PER_BLOCK;              // 4096
    const long long done        = full_blocks * ROWS_PER_BLOCK;
    const long long rem         = rows - done;

    if (full_blocks > 0) {
        bootstrap_mean_tile_kernel<<<dim3((unsigned)full_blocks), dim3(THREADS), 0, stream>>>(in, out);
    }
    if (rem > 0) {
        const unsigned tail_blocks = (unsigned)((rem + THREADS - 1) / THREADS);
        bootstrap_mean_tail_kernel<<<dim3(tail_blocks), dim3(THREADS), 0, stream>>>(in, out, done, rows);
    }
}